// PointPillarVoxelization_48850958025285
// MI455X (gfx1250) — compile-verified
//
#include <hip/hip_runtime.h>

namespace {

constexpr int   kNX   = 432;
constexpr int   kNY   = 496;
constexpr int   kCap  = kNX * kNY;        // 214272 (NZ == 1)
constexpr int   kMaxP = 100;

constexpr int kScanBlock = 256;
constexpr int kScanItems = 4;
constexpr int kScanTile  = kScanBlock * kScanItems;                 // 1024
constexpr int kNumScanBlocks = (kCap + kScanTile - 1) / kScanTile;  // 210
static_assert(kNumScanBlocks <= kScanBlock, "scanB assumes a single block");

// ---------------- gfx1250 async global<->LDS helpers (ASYNCcnt path) ----------------
// LDS byte offset = low 32 bits of the generic pointer (flat LDS addr = {aperture, off32}).
__device__ __forceinline__ unsigned lds_off(const void* p) {
  return (unsigned)(unsigned long long)(uintptr_t)p;
}
__device__ __forceinline__ void async_load_lds_b32(unsigned ldsOff, unsigned long long gaddr) {
  asm volatile("global_load_async_to_lds_b32 %0, %1, off"
               :: "v"(ldsOff), "v"(gaddr) : "memory");
}
__device__ __forceinline__ void async_store_lds_b128(unsigned long long gaddr, unsigned ldsOff) {
  asm volatile("global_store_async_from_lds_b128 %0, %1, off"
               :: "v"(gaddr), "v"(ldsOff) : "memory");
}
__device__ __forceinline__ void wait_async0() {
  asm volatile("s_wait_asynccnt 0x0" ::: "memory");
}
__device__ __forceinline__ void wait_ds0() {
  asm volatile("s_wait_dscnt 0x0" ::: "memory");
}

// ---------------------------------------------------------------- init
// Zero-fill 343MB of out_voxels by streaming async b128 stores from a zeroed LDS tile.
__global__ void k_init_voxels(float* __restrict__ outVox, long n4) {
  __shared__ float4 ztile[256];
  ztile[threadIdx.x] = make_float4(0.f, 0.f, 0.f, 0.f);
  wait_ds0();  // order LDS write before the async engine reads it
  const unsigned myLds = lds_off(&ztile[threadIdx.x]);
  long i      = (long)blockIdx.x * blockDim.x + threadIdx.x;
  long stride = (long)gridDim.x * blockDim.x;
  const unsigned long long base = (unsigned long long)(uintptr_t)outVox;
  for (; i < n4; i += stride) {
    async_store_lds_b128(base + (unsigned long long)i * 16ull, myLds);
  }
  wait_async0();
}

__global__ void k_init_small(int* __restrict__ cnt, int* __restrict__ cursor,
                             float* __restrict__ coords, float* __restrict__ counts) {
  int i = blockIdx.x * blockDim.x + threadIdx.x;
  if (i >= kCap) return;
  cnt[i]    = 0;
  cursor[i] = 0;
  coords[3 * i + 0] = -1.f;
  coords[3 * i + 1] = -1.f;
  coords[3 * i + 2] = -1.f;
  counts[i] = 0.f;
}

// ---------------------------------------------------------------- pass 1: lin + histogram
__global__ void k_points(const float4* __restrict__ pts, int* __restrict__ lins,
                         int* __restrict__ cnt, int n) {
  int i = blockIdx.x * blockDim.x + threadIdx.x;
  if (i >= n) return;
  __builtin_prefetch(pts + i + 4096, 0, 0);  // gfx1250 global_prefetch_b8 (speculative)
  float4 p = pts[i];
  bool valid = (p.x >= 0.0f)    && (p.x < 69.12f) &&
               (p.y >= -39.68f) && (p.y < 39.68f) &&
               (p.z >= -3.0f)   && (p.z < 1.0f);
  int lin = kCap;
  if (valid) {
    // mirror reference: (p - rmin) / vsize, floorf, clip; z clips to 0 so lin = y*NX + x
    int ix = (int)floorf((p.x - 0.0f)      / 0.16f);
    int iy = (int)floorf((p.y - (-39.68f)) / 0.16f);
    ix = min(max(ix, 0), kNX - 1);
    iy = min(max(iy, 0), kNY - 1);
    lin = iy * kNX + ix;
    atomicAdd(&cnt[lin], 1);
  }
  lins[i] = lin;
}

// ---------------------------------------------------------------- pass 2: occupancy scan
__global__ void k_scanA(const int* __restrict__ cnt, int* __restrict__ blockSums) {
  __shared__ int sm[kScanBlock];
  int b = blockIdx.x, t = threadIdx.x;
  int base = b * kScanTile + t * kScanItems;
  int s = 0;
#pragma unroll
  for (int k = 0; k < kScanItems; ++k) {
    int v = base + k;
    if (v < kCap) s += (cnt[v] > 0);
  }
  sm[t] = s;
  __syncthreads();
  for (int off = kScanBlock / 2; off > 0; off >>= 1) {
    if (t < off) sm[t] += sm[t + off];
    __syncthreads();
  }
  if (t == 0) blockSums[b] = sm[0];
}

__global__ void k_scanB(const int* __restrict__ blockSums, int* __restrict__ blockOffs,
                        int* __restrict__ nOcc) {
  __shared__ int sm[kScanBlock];
  int t = threadIdx.x;
  int v = (t < kNumScanBlocks) ? blockSums[t] : 0;
  sm[t] = v;
  __syncthreads();
  for (int off = 1; off < kScanBlock; off <<= 1) {
    int add = (t >= off) ? sm[t - off] : 0;
    __syncthreads();
    sm[t] += add;
    __syncthreads();
  }
  if (t < kNumScanBlocks) blockOffs[t] = sm[t] - v;  // exclusive
  if (t == kScanBlock - 1) *nOcc = sm[t];
}

__global__ void k_scanC(const int* __restrict__ cnt, const int* __restrict__ blockOffs,
                        int* __restrict__ rowArr, int* __restrict__ rowCnt,
                        float* __restrict__ coords, float* __restrict__ counts) {
  __shared__ int sm[kScanBlock];
  int b = blockIdx.x, t = threadIdx.x;
  int base = b * kScanTile + t * kScanItems;
  int cvals[kScanItems];
  int occ[kScanItems];
  int s = 0;
#pragma unroll
  for (int k = 0; k < kScanItems; ++k) {
    int v = base + k;
    int c = (v < kCap) ? cnt[v] : 0;
    cvals[k] = c;
    occ[k] = (c > 0);
    s += occ[k];
  }
  sm[t] = s;
  __syncthreads();
  for (int off = 1; off < kScanBlock; off <<= 1) {
    int add = (t >= off) ? sm[t - off] : 0;
    __syncthreads();
    sm[t] += add;
    __syncthreads();
  }
  int r = blockOffs[b] + (sm[t] - s);  // exclusive prefix for this thread
#pragma unroll
  for (int k = 0; k < kScanItems; ++k) {
    if (occ[k]) {
      int v = base + k;
      rowArr[v] = r;
      rowCnt[r] = cvals[k];
      int y = v / kNX;
      int x = v - y * kNX;
      coords[3 * r + 0] = 0.f;        // z (NZ==1)
      coords[3 * r + 1] = (float)y;
      coords[3 * r + 2] = (float)x;
      counts[r] = (float)cvals[k];
      ++r;
    }
  }
}

// ---------------------------------------------------------------- pass 3: ticket scatter (point index into slot comp0)
__global__ void k_scatter(const int* __restrict__ lins, const int* __restrict__ rowArr,
                          int* __restrict__ cursor, float* __restrict__ outVox, int n) {
  int i = blockIdx.x * blockDim.x + threadIdx.x;
  if (i >= n) return;
  int lin = lins[i];
  if (lin >= kCap) return;
  int t = atomicAdd(&cursor[lin], 1);
  if (t < kMaxP) {
    int r = rowArr[lin];
    ((int*)outVox)[((size_t)r * kMaxP + t) * 4] = i;  // comps 1..3 stay zero from init
  }
}

// ---------------------------------------------------------------- pass 4: per-voxel stable repair + feature write
__global__ void k_repair(const float4* __restrict__ pts, const int* __restrict__ rowCnt,
                         const int* __restrict__ nOccPtr, float* __restrict__ outVox) {
  int r = blockIdx.x;
  if (r >= *nOccPtr) return;
  int k = rowCnt[r];
  if (k > kMaxP) k = kMaxP;
  __shared__ int sIdx[kMaxP];
  int t = threadIdx.x;
  if (t < k) {
    // async gather of strided slot indices straight into LDS (no VGPR round-trip)
    unsigned long long g =
        (unsigned long long)(uintptr_t)(outVox + ((size_t)r * kMaxP + t) * 4);
    async_load_lds_b32(lds_off(&sIdx[t]), g);
  }
  wait_async0();       // this wave's gathers have landed in LDS
  __syncthreads();     // cross-wave visibility within the block
  if (t == 0) {  // restore original-index (stable) order; k is tiny (Poisson ~1.2)
    for (int a = 1; a < k; ++a) {
      int key = sIdx[a];
      int b = a - 1;
      while (b >= 0 && sIdx[b] > key) { sIdx[b + 1] = sIdx[b]; --b; }
      sIdx[b + 1] = key;
    }
  }
  __syncthreads();
  if (t < k) {
    float4 p = pts[sIdx[t]];
    *(float4*)(outVox + ((size_t)r * kMaxP + t) * 4) = p;
  }
}

}  // namespace

extern "C" void kernel_launch(void* const* d_in, const int* in_sizes, int n_in,
                              void* d_out, int out_size, void* d_ws, size_t ws_size,
                              hipStream_t stream) {
  (void)n_in; (void)out_size; (void)ws_size;
  const float4* pts = (const float4*)d_in[0];
  const int n = in_sizes[0] / 4;

  float* outVox    = (float*)d_out;                          // [CAP,100,4]
  float* outCoords = outVox + (size_t)kCap * kMaxP * 4;      // [CAP,3]
  float* outCounts = outCoords + (size_t)kCap * 3;           // [CAP]

  int* ws        = (int*)d_ws;
  int* lins      = ws;                    // [n]
  int* cnt       = lins + n;              // [CAP]
  int* cursor    = cnt + kCap;            // [CAP]
  int* rowArr    = cursor + kCap;         // [CAP]
  int* rowCnt    = rowArr + kCap;         // [CAP]
  int* blockSums = rowCnt + kCap;         // [kNumScanBlocks]
  int* blockOffs = blockSums + kNumScanBlocks;
  int* nOcc      = blockOffs + kNumScanBlocks;

  const long n4 = (long)kCap * kMaxP;  // float4 elements of out_voxels
  k_init_voxels<<<4096, 256, 0, stream>>>(outVox, n4);
  k_init_small<<<(kCap + 255) / 256, 256, 0, stream>>>(cnt, cursor, outCoords, outCounts);

  k_points<<<(n + 255) / 256, 256, 0, stream>>>(pts, lins, cnt, n);

  k_scanA<<<kNumScanBlocks, kScanBlock, 0, stream>>>(cnt, blockSums);
  k_scanB<<<1, kScanBlock, 0, stream>>>(blockSums, blockOffs, nOcc);
  k_scanC<<<kNumScanBlocks, kScanBlock, 0, stream>>>(cnt, blockOffs, rowArr, rowCnt,
                                                     outCoords, outCounts);

  k_scatter<<<(n + 255) / 256, 256, 0, stream>>>(lins, rowArr, cursor, outVox, n);
  k_repair<<<kCap, 128, 0, stream>>>(pts, rowCnt, nOcc, outVox);
}